// sclayer_58746562675072
// MI455X (gfx1250) — compile-verified
//
#include <hip/hip_runtime.h>
#include <stdint.h>

// ---------------------------------------------------------------------------
// Reference: out[b, 0:9]  = [x, cos(x), sin(x)]           (3 input channels)
//            out[b, 9:27] = base[II[p]] * base[JJ[p]]     (18 fixed pairs)
// Shapes: in  [32, 3, 512, 512] f32   -> out [32, 27, 512, 512] f32
//
// MI455X roofline: ~96 MB in + ~864 MB out = ~0.96 GB @ 23.3 TB/s => ~41 us.
// Pure streaming kernel; no matrix structure => WMMA inapplicable.
// CDNA5 paths used: global_load_async_to_lds_b128 double-buffer pipeline,
// s_wait_asynccnt, ds_load_b128 read-back, v_sin/cos_f32, NT b128 stores.
// ---------------------------------------------------------------------------

typedef float v4f __attribute__((ext_vector_type(4)));

#define HWSZ            (512 * 512)                 // 262144 pixels / plane
#define BATCH           32
#define TPB             256                         // 8 wave32s / block
#define PIX_PER_THREAD  4                           // one b128 per channel
#define TILE_PIX        (TPB * PIX_PER_THREAD)      // 1024 px / tile
#define NTILES          ((BATCH * HWSZ) / TILE_PIX) // 8192 tiles
#define BLOCKS          2048
#define TILES_PER_BLOCK (NTILES / BLOCKS)           // 4
#define TILES_PER_PLANE (HWSZ / TILE_PIX)           // 256

// Generic (__shared__) pointer -> 32-bit LDS byte offset for async-DMA VDST.
__device__ __forceinline__ uint32_t lds_off_u32(const void* p) {
  return (uint32_t)(uintptr_t)(__attribute__((address_space(3))) const void*)p;
}

// Async DMA: 16 bytes per lane, global -> LDS, tracked by ASYNCcnt.
__device__ __forceinline__ void async_load_b128(uint32_t lds_byte_off,
                                                const float* gptr) {
  asm volatile("global_load_async_to_lds_b128 %0, %1, off"
               :
               : "v"(lds_byte_off), "v"((uint64_t)(uintptr_t)gptr)
               : "memory");
}

__global__ __launch_bounds__(TPB) void sclayer_kernel(
    const float* __restrict__ in, float* __restrict__ out) {
  // Double-buffered staging: 2 x 3 channels x 1024 px x 4 B = 24 KB LDS.
  __shared__ float tile[2][3][TILE_PIX];

  const int tid = threadIdx.x;
  const uint32_t lds0 = lds_off_u32(&tile[0][0][0]);

  // Issue the 3 per-lane async b128 loads (one per input channel) for tile t.
  auto issue = [&](int t, int buf) {
    const int b = t / TILES_PER_PLANE;
    const int s = (t % TILES_PER_PLANE) * TILE_PIX + tid * PIX_PER_THREAD;
    const float* gbase = in + (size_t)b * 3 * HWSZ + s;
    const uint32_t loff = lds0 + (uint32_t)buf * (3 * TILE_PIX * 4) + tid * 16;
#pragma unroll
    for (int c = 0; c < 3; ++c)
      async_load_b128(loff + c * (TILE_PIX * 4), gbase + (size_t)c * HWSZ);
  };

  // Pair index tables (exactly the reference enumeration order).
  const int II[18] = {0, 0, 0, 1, 1, 1, 2, 2, 2, 3, 3, 4, 4, 5, 5, 6, 7, 8};
  const int JJ[18] = {0, 3, 6, 1, 4, 7, 2, 5, 8, 3, 6, 4, 7, 5, 8, 6, 7, 8};

  issue(blockIdx.x, 0);  // prologue: prefetch first tile

  for (int k = 0; k < TILES_PER_BLOCK; ++k) {
    const int t = blockIdx.x + k * BLOCKS;
    const int buf = k & 1;

    if (k + 1 < TILES_PER_BLOCK) {
      // Prefetch next tile into the other buffer, then wait only for the
      // current tile's 3 loads (in-order completion => ASYNCcnt <= 3).
      issue(blockIdx.x + (k + 1) * BLOCKS, buf ^ 1);
      asm volatile("s_wait_asynccnt 0x3" ::: "memory");
    } else {
      asm volatile("s_wait_asynccnt 0x0" ::: "memory");
    }
    // Each lane reads back exactly the LDS bytes its own async loads wrote,
    // so no workgroup barrier is required — only the ASYNCcnt wait above.

    v4f bb[9];
#pragma unroll
    for (int c = 0; c < 3; ++c) {
      v4f x = *(const v4f*)&tile[buf][c][tid * PIX_PER_THREAD];  // ds_load_b128
      v4f cv, sv;
#pragma unroll
      for (int e = 0; e < 4; ++e) {
        cv[e] = __cosf(x[e]);  // v_cos_f32 (trans, co-executes with VALU)
        sv[e] = __sinf(x[e]);  // v_sin_f32
      }
      bb[c] = x;
      bb[3 + c] = cv;
      bb[6 + c] = sv;
    }

    const int b = t / TILES_PER_PLANE;
    const int s = (t % TILES_PER_PLANE) * TILE_PIX + tid * PIX_PER_THREAD;
    float* op = out + (size_t)b * 27 * HWSZ + s;

    // 27 aligned non-temporal b128 stores: write-once stream, keep caches clean.
#pragma unroll
    for (int c = 0; c < 9; ++c)
      __builtin_nontemporal_store(bb[c], (v4f*)(op + (size_t)c * HWSZ));
#pragma unroll
    for (int p = 0; p < 18; ++p)
      __builtin_nontemporal_store(bb[II[p]] * bb[JJ[p]],
                                  (v4f*)(op + (size_t)(9 + p) * HWSZ));
  }
}

extern "C" void kernel_launch(void* const* d_in, const int* in_sizes, int n_in,
                              void* d_out, int out_size, void* d_ws,
                              size_t ws_size, hipStream_t stream) {
  (void)in_sizes; (void)n_in; (void)out_size; (void)d_ws; (void)ws_size;
  const float* in = (const float*)d_in[0];
  float* out = (float*)d_out;
  sclayer_kernel<<<BLOCKS, TPB, 0, stream>>>(in, out);
}